// StandardSAE_27745488732933
// MI455X (gfx1250) — compile-verified
//
#include <hip/hip_runtime.h>
#include <hip/hip_bf16.h>

// ---------------------------------------------------------------------------
// Top-K Sparse Autoencoder forward for MI455X (gfx1250, wave32).
//
//   pre_codes = x @ W_enc^T + b_enc        [8192 x 16384]   (fp32 WMMA GEMM)
//   topk/scatter -> codes                  [8192 x 16384]   (LDS radix-select)
//   reconstruction = codes @ D             [8192 x 768]     (sparse gather-sum)
//
// HBM-store-bound (~1.6 GB traffic); GEMM uses V_WMMA_F32_16X16X4_F32 to stay
// bit-faithful to the fp32 reference (top-k index selection is sensitive).
// ---------------------------------------------------------------------------

typedef float v2f __attribute__((ext_vector_type(2)));
typedef float v8f __attribute__((ext_vector_type(8)));

#define N_ROWS   8192
#define D_MODEL  768
#define N_FEAT   16384
#define TOP_K    64

// ============================ Kernel 1: encoder GEMM ========================
// Block = 128 threads (4 waves). Wave w computes rows [m0+32w, m0+32w+32),
// cols [n0, n0+64) as a 2(M) x 4(N) grid of 16x16 fp32 WMMA tiles, K stepped
// by 4 (V_WMMA_F32_16X16X4_F32).
//
// 32-bit A 16x4 lane layout: lanes 0-15 hold {K=0,K=1} for M=lane,
//                            lanes 16-31 hold {K=2,K=3} for M=lane-16.
// 32-bit B 4x16 lane layout: lanes 0-15 hold {K=0,K=1} for N=lane,
//                            lanes 16-31 hold {K=2,K=3} for N=lane-16.
// Both map to contiguous float2 loads from row-major x / W_enc.
__global__ __launch_bounds__(128)
void sae_encode_gemm(const float* __restrict__ x,
                     const float* __restrict__ W,
                     const float* __restrict__ bias,
                     float* __restrict__ pre)
{
    const int lane = threadIdx.x & 31;
    const int wave = threadIdx.x >> 5;
    const int n0   = blockIdx.x * 64;
    const int m0   = blockIdx.y * 128 + wave * 32;
    const int half = lane >> 4;          // 0: K=0..1, 1: K=2..3
    const int lr   = lane & 15;
    const int koff = half * 2;

    const float* xp0 = x + (size_t)(m0 + lr) * D_MODEL + koff;        // M tile 0
    const float* xp1 = xp0 + (size_t)16 * D_MODEL;                    // M tile 1
    const float* wp  = W + (size_t)(n0 + lr) * D_MODEL + koff;        // N tile 0

    v8f acc[2][4];
#pragma unroll
    for (int mi = 0; mi < 2; ++mi)
#pragma unroll
        for (int ni = 0; ni < 4; ++ni)
            acc[mi][ni] = (v8f){};

    for (int k = 0; k < D_MODEL; k += 4) {
        v2f a0 = *(const v2f*)(xp0 + k);
        v2f a1 = *(const v2f*)(xp1 + k);
#pragma unroll
        for (int ni = 0; ni < 4; ++ni) {
            v2f bb = *(const v2f*)(wp + (size_t)ni * 16 * D_MODEL + k);
            acc[0][ni] = __builtin_amdgcn_wmma_f32_16x16x4_f32(
                false, a0, false, bb, (short)0, acc[0][ni], false, false);
            acc[1][ni] = __builtin_amdgcn_wmma_f32_16x16x4_f32(
                false, a1, false, bb, (short)0, acc[1][ni], false, false);
        }
    }

    // C/D layout: lanes 0-15 -> N=lane, M=vgpr(0..7); lanes 16-31 -> N=lane-16, M=8+vgpr.
#pragma unroll
    for (int ni = 0; ni < 4; ++ni) {
        const int n = n0 + ni * 16 + lr;
        const float bv = bias[n];
#pragma unroll
        for (int mi = 0; mi < 2; ++mi) {
            const int mbase = m0 + mi * 16 + half * 8;
#pragma unroll
            for (int r = 0; r < 8; ++r) {
                pre[(size_t)(mbase + r) * N_FEAT + n] = acc[mi][ni][r] + bv;
            }
        }
    }
}

// ======================= Kernel 2: top-k + dense scatter ====================
__device__ __forceinline__ unsigned int f2key(float v) {
    unsigned int u = __float_as_uint(v);
    return u ^ ((unsigned int)((int)u >> 31) | 0x80000000u);   // order-preserving
}
__device__ __forceinline__ float key2f(unsigned int k) {
    unsigned int u = (k & 0x80000000u) ? (k ^ 0x80000000u) : ~k;
    return __uint_as_float(u);
}

// One workgroup per row. Whole row (64 KB) staged in LDS; 4-pass byte radix
// select finds the exact TOP_K-th largest key; ties broken by lowest index
// (matches jax.lax.top_k). Deterministic: atomics are commutative counts only.
__global__ __launch_bounds__(256)
void sae_topk_scatter(const float* __restrict__ pre,
                      float* __restrict__ codes,
                      float* __restrict__ tv,     // [N_ROWS][TOP_K] values
                      int*   __restrict__ ti)     // [N_ROWS][TOP_K] indices
{
    extern __shared__ unsigned int smem[];
    unsigned int* keys  = smem;               // 16384
    unsigned int* hist  = smem + 16384;       // 256
    unsigned int* sgt   = smem + 16640;       // 256
    unsigned int* seq   = smem + 16896;       // 256
    unsigned int* maskw = smem + 17152;       // 512 (selection bitmask)
    unsigned int* bc    = smem + 17664;       // 4: [pfx, pmask, remaining]

    const int row = blockIdx.x;
    const int tid = threadIdx.x;
    const float* prow = pre + (size_t)row * N_FEAT;

    for (int i = tid; i < N_FEAT; i += 256)
        keys[i] = f2key(prow[i]);
    if (tid == 0) { bc[0] = 0u; bc[1] = 0u; bc[2] = TOP_K; }
    __syncthreads();

    // ---- radix select: find key of the TOP_K-th largest element ----
    for (int p = 0; p < 4; ++p) {
        const int shift = 24 - 8 * p;
        hist[tid] = 0u;
        __syncthreads();
        const unsigned int pfx = bc[0], pmask = bc[1];
        for (int i = tid; i < N_FEAT; i += 256) {
            const unsigned int kk = keys[i];
            if ((kk & pmask) == pfx)
                atomicAdd(&hist[(kk >> shift) & 255u], 1u);
        }
        __syncthreads();
        if (tid == 0) {
            unsigned int rem = bc[2], run = 0u, chosen = 0u;
            for (int b = 255; b >= 0; --b) {
                const unsigned int c = hist[b];
                if (run + c >= rem) { chosen = (unsigned int)b; break; }
                run += c;
            }
            bc[0] = pfx | (chosen << shift);
            bc[1] = pmask | (0xFFu << shift);
            bc[2] = rem - run;                 // still-needed among == prefix
        }
        __syncthreads();
    }
    const unsigned int thr  = bc[0];   // exact key of K-th largest
    const unsigned int need = bc[2];   // how many ==thr to keep (lowest index first)

    // ---- exclusive prefix of (>thr) and (==thr) counts, index order ----
    {
        unsigned int cg = 0u, ce = 0u;
        const int base = tid * 64;
        for (int j = 0; j < 64; ++j) {
            const unsigned int kk = keys[base + j];
            cg += (kk > thr);
            ce += (kk == thr);
        }
        sgt[tid] = cg; seq[tid] = ce;
    }
    __syncthreads();
    if (tid == 0) {
        unsigned int g = 0u, e = 0u;
        for (int t = 0; t < 256; ++t) {
            const unsigned int tg = sgt[t], te = seq[t];
            sgt[t] = g; seq[t] = e;
            g += tg; e += te;
        }
    }
    __syncthreads();

    // ---- ordered selection: compact output position == rank by index ----
    {
        unsigned int run_gt = sgt[tid];
        unsigned int run_eq = seq[tid];
        unsigned int m0b = 0u, m1b = 0u;
        const int base = tid * 64;
        for (int j = 0; j < 64; ++j) {
            const int i = base + j;
            const unsigned int kk = keys[i];
            bool sel = false;
            unsigned int pos = 0u;
            if (kk > thr) {
                pos = run_gt + (run_eq < need ? run_eq : need);
                sel = true; ++run_gt;
            } else if (kk == thr) {
                if (run_eq < need) { pos = run_gt + run_eq; sel = true; }
                ++run_eq;
            }
            if (sel) {
                tv[(size_t)row * TOP_K + pos] = key2f(kk);
                ti[(size_t)row * TOP_K + pos] = i;
                if (j < 32) m0b |= 1u << j; else m1b |= 1u << (j - 32);
            }
        }
        maskw[2 * tid]     = m0b;
        maskw[2 * tid + 1] = m1b;
    }
    __syncthreads();

    // ---- coalesced dense codes row write ----
    float* crow = codes + (size_t)row * N_FEAT;
    for (int i = tid; i < N_FEAT; i += 256) {
        const bool sel = (maskw[i >> 5] >> (i & 31)) & 1u;
        crow[i] = sel ? key2f(keys[i]) : 0.0f;
    }
}

// ====================== Kernel 3: sparse decode (gather) ====================
__global__ __launch_bounds__(256)
void sae_decode(const float* __restrict__ tv,
                const int*   __restrict__ ti,
                const float* __restrict__ D,
                float* __restrict__ recon)
{
    __shared__ float sv[TOP_K];
    __shared__ int   si[TOP_K];
    const int row = blockIdx.x;
    if (threadIdx.x < TOP_K) {
        sv[threadIdx.x] = tv[(size_t)row * TOP_K + threadIdx.x];
        si[threadIdx.x] = ti[(size_t)row * TOP_K + threadIdx.x];
    }
    __syncthreads();
    for (int d = threadIdx.x; d < D_MODEL; d += 256) {
        float acc = 0.0f;
#pragma unroll
        for (int k = 0; k < TOP_K; ++k)
            acc += sv[k] * D[(size_t)si[k] * D_MODEL + d];
        recon[(size_t)row * D_MODEL + d] = acc;
    }
}

// ================================ launcher ==================================
extern "C" void kernel_launch(void* const* d_in, const int* in_sizes, int n_in,
                              void* d_out, int out_size, void* d_ws, size_t ws_size,
                              hipStream_t stream) {
    const float* x    = (const float*)d_in[0];   // [8192, 768]
    const float* Wenc = (const float*)d_in[1];   // [16384, 768]
    const float* benc = (const float*)d_in[2];   // [16384]
    const float* Dict = (const float*)d_in[3];   // [16384, 768]

    // d_out = reconstruction | codes | pre_codes (flat, return order)
    float* recon = (float*)d_out;
    float* codes = recon + (size_t)N_ROWS * D_MODEL;
    float* pre   = codes + (size_t)N_ROWS * N_FEAT;

    // workspace: compact top-k (values then indices), 4 MB
    float* tv = (float*)d_ws;
    int*   ti = (int*)(tv + (size_t)N_ROWS * TOP_K);

    // 1) encoder GEMM: 128N x 64? -> grid (N/64, M/128), 4 waves/block
    dim3 ggrid(N_FEAT / 64, N_ROWS / 128);
    sae_encode_gemm<<<ggrid, 128, 0, stream>>>(x, Wenc, benc, pre);

    // 2) per-row top-k + scatter; dynamic LDS = 17668 dwords (~69 KB < 320 KB/WGP)
    const size_t smem_bytes = (size_t)(16384 + 256 + 256 + 256 + 512 + 4) * sizeof(unsigned int);
    sae_topk_scatter<<<N_ROWS, 256, smem_bytes, stream>>>(pre, codes, tv, ti);

    // 3) sparse decode
    sae_decode<<<N_ROWS, 256, 0, stream>>>(tv, ti, Dict, recon);
}